// GCN_att_v2_67937792688718
// MI455X (gfx1250) — compile-verified
//
#include <hip/hip_runtime.h>

typedef __attribute__((ext_vector_type(2))) float v2f;
typedef __attribute__((ext_vector_type(8))) float v8f;

// ---------------------------------------------------------------------------
// Degree / normalization kernels
// ---------------------------------------------------------------------------
__global__ void k_deg_init(float* __restrict__ deg, float* __restrict__ pooled, int n) {
    int i = blockIdx.x * blockDim.x + threadIdx.x;
    if (i < n) deg[i] = 1.0f;           // self-loop contributes +1 to degree
    if (i < 16) pooled[i] = 0.0f;       // zero the pooling accumulator
}

__global__ void k_deg_accum(const int* __restrict__ dst, float* __restrict__ deg, int e) {
    int i = blockIdx.x * blockDim.x + threadIdx.x;
    if (i < e) atomicAdd(&deg[dst[i]], 1.0f);
}

__global__ void k_rsqrt_inplace(float* __restrict__ deg, int n) {
    int i = blockIdx.x * blockDim.x + threadIdx.x;
    if (i < n) deg[i] = rsqrtf(deg[i]); // deg buffer becomes dinv
}

// ---------------------------------------------------------------------------
// Dense GEMM via V_WMMA_F32_16X16X4_F32 (fp32 in, fp32 acc — matches reference
// precision). One wave computes a 16-row strip x FOUT cols. A-fragment layout
// per ISA 7.12.2 (32-bit A 16x4): lanes 0-15 hold K=0,1; lanes 16-31 hold
// K=2,3. C/D: VGPR r -> M = r + 8*(lane/16), N = lane%16.
// RELU is a template parameter so the hot loop has no cndmask noise.
// ---------------------------------------------------------------------------
template <int FOUT, bool RELU>
__global__ void k_gemm_wmma(const float* __restrict__ X, const float* __restrict__ W,
                            float* __restrict__ Y, int Fin, int tilesM) {
    const int wave = blockIdx.x * (blockDim.x >> 5) + (threadIdx.x >> 5);
    if (wave >= tilesM) return;                 // wave-uniform guard (EXEC stays all-1s)
    const int lane = threadIdx.x & 31;
    const int row  = lane & 15;                 // M (for A) / N (for B,C)
    const int half = lane >> 4;                 // K-half selector
    const int m0   = wave << 4;
    constexpr int NT = FOUT >> 4;               // 16-col tiles (<= 4)

    v8f acc[NT] = {};

    const float* xrow = X + (size_t)(m0 + row) * Fin + 2 * half;
    for (int k0 = 0; k0 < Fin; k0 += 4) {
        float ax = xrow[k0];
        float ay = xrow[k0 + 1];
        if (RELU) { ax = fmaxf(ax, 0.0f); ay = fmaxf(ay, 0.0f); }
        v2f a; a.x = ax; a.y = ay;
        const float* wrow = W + (size_t)(k0 + 2 * half) * FOUT + row;
#pragma unroll
        for (int nt = 0; nt < NT; ++nt) {
            v2f b;
            b.x = wrow[nt * 16];
            b.y = wrow[nt * 16 + FOUT];
            acc[nt] = __builtin_amdgcn_wmma_f32_16x16x4_f32(
                false, a, false, b, (short)0, acc[nt], false, false);
        }
    }
#pragma unroll
    for (int nt = 0; nt < NT; ++nt) {
        float* yp = Y + (size_t)(m0 + 8 * half) * FOUT + nt * 16 + row;
#pragma unroll
        for (int r = 0; r < 8; ++r)
            yp[(size_t)r * FOUT] = acc[nt][r];
    }
}

// ---------------------------------------------------------------------------
// agg = xw * self_w + bias  (initializes the scatter target; no memset needed)
// ---------------------------------------------------------------------------
__global__ void k_init_agg(const float* __restrict__ xw, const float* __restrict__ dinv,
                           const float* __restrict__ bias, float* __restrict__ agg,
                           long long total, int fmask, int fshift) {
    long long i = (long long)blockIdx.x * blockDim.x + threadIdx.x;
    if (i < total) {
        int node = (int)(i >> fshift);
        int f    = (int)i & fmask;
        float d  = dinv[node];
        agg[i] = xw[i] * (d * d) + bias[f];
    }
}

// ---------------------------------------------------------------------------
// Edge scatter: agg[dst] += xw[src] * dinv[src]*dinv[dst]
// One thread per float4 chunk of one edge's message: the src/dst/dinv loads
// are amortized 4x, the gather becomes global_load_b128, and the adds stay
// per-float global_atomic_add_f32 (no packed f32 atomic exists). Both xw and
// agg buffers (<= 25.6 MB each) are L2-resident on the 192 MB L2.
// ---------------------------------------------------------------------------
template <int FSHIFT>   // log2(F); chunks per edge = F/4 = 1 << (FSHIFT-2)
__global__ void k_scatter4(const float* __restrict__ xw, const int* __restrict__ src,
                           const int* __restrict__ dst, const float* __restrict__ dinv,
                           float* __restrict__ agg, long long total4) {
    constexpr int QSHIFT = FSHIFT - 2;
    constexpr int QMASK  = (1 << QSHIFT) - 1;
    const long long stride = (long long)gridDim.x * blockDim.x;
    for (long long t = (long long)blockIdx.x * blockDim.x + threadIdx.x;
         t < total4; t += stride) {
        int e = (int)(t >> QSHIFT);
        int q = ((int)t & QMASK) << 2;          // feature offset (multiple of 4)
        int s = src[e];
        int d = dst[e];
        float w = dinv[s] * dinv[d];
        const float4 v = *(const float4*)(xw + ((size_t)s << FSHIFT) + q);
        float* ap = agg + ((size_t)d << FSHIFT) + q;
        atomicAdd(ap + 0, v.x * w);
        atomicAdd(ap + 1, v.y * w);
        atomicAdd(ap + 2, v.z * w);
        atomicAdd(ap + 3, v.w * w);
    }
}

// ---------------------------------------------------------------------------
// Mean pool (with final ReLU fused): pooled[f] += sum_i relu(h[i][f])
// F3 == 16; thread's feature lane is invariant under the grid stride.
// ---------------------------------------------------------------------------
__global__ void k_pool(const float* __restrict__ h, float* __restrict__ pooled,
                       long long total) {
    __shared__ float s[16];
    if (threadIdx.x < 16) s[threadIdx.x] = 0.0f;
    __syncthreads();
    const long long stride = (long long)gridDim.x * blockDim.x;
    float acc = 0.0f;
    const int f = threadIdx.x & 15;
    for (long long t = (long long)blockIdx.x * blockDim.x + threadIdx.x;
         t < total; t += stride)
        acc += fmaxf(h[t], 0.0f);
    atomicAdd(&s[f], acc);
    __syncthreads();
    if (threadIdx.x < 16) atomicAdd(&pooled[threadIdx.x], s[threadIdx.x]);
}

// ---------------------------------------------------------------------------
// Tiny head: mean -> 16x8 MLP (relu) -> 8x10 -> log_softmax. 208 MACs total.
// ---------------------------------------------------------------------------
__global__ void k_head(const float* __restrict__ pooled,
                       const float* __restrict__ Wf, const float* __restrict__ bf,
                       const float* __restrict__ Ws, const float* __restrict__ bs,
                       float* __restrict__ out, float invN) {
    if (threadIdx.x != 0 || blockIdx.x != 0) return;
    float p[16], hid[8], sc[10];
    for (int f = 0; f < 16; ++f) p[f] = pooled[f] * invN;
    for (int j = 0; j < 8; ++j) {
        float a = bf[j];
        for (int f = 0; f < 16; ++f) a += p[f] * Wf[f * 8 + j];
        hid[j] = fmaxf(a, 0.0f);
    }
    float m = -1e30f;
    for (int c = 0; c < 10; ++c) {
        float a = bs[c];
        for (int j = 0; j < 8; ++j) a += hid[j] * Ws[j * 10 + c];
        sc[c] = a;
        m = fmaxf(m, a);
    }
    float se = 0.0f;
    for (int c = 0; c < 10; ++c) se += expf(sc[c] - m);
    float lse = m + logf(se);
    for (int c = 0; c < 10; ++c) out[c] = sc[c] - lse;
}

// ---------------------------------------------------------------------------
// Host-side launch
// ---------------------------------------------------------------------------
extern "C" void kernel_launch(void* const* d_in, const int* in_sizes, int n_in,
                              void* d_out, int out_size, void* d_ws, size_t ws_size,
                              hipStream_t stream) {
    const float* feat = (const float*)d_in[0];
    const int*   eidx = (const int*)d_in[1];
    const float* W1 = (const float*)d_in[2];  const float* b1 = (const float*)d_in[3];
    const float* W2 = (const float*)d_in[4];  const float* b2 = (const float*)d_in[5];
    const float* W3 = (const float*)d_in[6];  const float* b3 = (const float*)d_in[7];
    const float* Wf = (const float*)d_in[8];  const float* bf = (const float*)d_in[9];
    const float* Ws = (const float*)d_in[10]; const float* bs = (const float*)d_in[11];

    const int N = in_sizes[0] / 256;   // 100000
    const int E = in_sizes[1] / 2;     // 3200000
    const int* src = eidx;
    const int* dst = eidx + E;

    // Workspace layout (floats): dinv[N] | bufA[N*64] | bufB[N*64] | pooled[16]
    float* ws     = (float*)d_ws;
    float* dinv   = ws;
    float* bufA   = dinv + N;                      // xw per layer
    float* bufB   = bufA + (size_t)N * 64;         // agg / next-layer input
    float* pooled = bufB + (size_t)N * 64;

    const int T = 256;

    // --- normalization -----------------------------------------------------
    k_deg_init<<<(N + T - 1) / T, T, 0, stream>>>(dinv, pooled, N);
    k_deg_accum<<<(E + T - 1) / T, T, 0, stream>>>(dst, dinv, E);
    k_rsqrt_inplace<<<(N + T - 1) / T, T, 0, stream>>>(dinv, N);

    const int tilesM = N / 16;                     // 6250
    const int gemmBlocks = (tilesM + 7) / 8;       // 8 waves per 256-thread block

    // --- layer 1: 256 -> 64 ------------------------------------------------
    k_gemm_wmma<64, false><<<gemmBlocks, T, 0, stream>>>(feat, W1, bufA, 256, tilesM);
    {
        long long total = (long long)N * 64;
        k_init_agg<<<(int)((total + T - 1) / T), T, 0, stream>>>(bufA, dinv, b1, bufB, total, 63, 6);
        k_scatter4<6><<<4096, T, 0, stream>>>(bufA, src, dst, dinv, bufB, (long long)E * 16);
    }

    // --- layer 2: 64 -> 32 (ReLU fused into A-load) ------------------------
    k_gemm_wmma<32, true><<<gemmBlocks, T, 0, stream>>>(bufB, W2, bufA, 64, tilesM);
    {
        long long total = (long long)N * 32;
        k_init_agg<<<(int)((total + T - 1) / T), T, 0, stream>>>(bufA, dinv, b2, bufB, total, 31, 5);
        k_scatter4<5><<<4096, T, 0, stream>>>(bufA, src, dst, dinv, bufB, (long long)E * 8);
    }

    // --- layer 3: 32 -> 16 -------------------------------------------------
    k_gemm_wmma<16, true><<<gemmBlocks, T, 0, stream>>>(bufB, W3, bufA, 32, tilesM);
    {
        long long total = (long long)N * 16;
        k_init_agg<<<(int)((total + T - 1) / T), T, 0, stream>>>(bufA, dinv, b3, bufB, total, 15, 4);
        k_scatter4<4><<<4096, T, 0, stream>>>(bufA, src, dst, dinv, bufB, (long long)E * 4);
    }

    // --- pool (ReLU fused) + head ------------------------------------------
    k_pool<<<1024, T, 0, stream>>>(bufB, pooled, (long long)N * 16);
    k_head<<<1, 32, 0, stream>>>(pooled, Wf, bf, Ws, bs, (float*)d_out, 1.0f / (float)N);
}